// GCNEncoder_77644418777891
// MI455X (gfx1250) — compile-verified
//
#include <hip/hip_runtime.h>

#define B_ 32
#define N_ 1000
#define D_ 256
#define L_ 3
#define S_ 16          // N-dimension reduction splits for kernel 1
#define HSTRIDE 260    // padded LDS row stride (260 % 64 == 4, odd multiple avoids 16-way conflicts)

typedef __attribute__((ext_vector_type(2))) float v2f;
typedef __attribute__((ext_vector_type(8))) float v8f;

// ---------------------------------------------------------------------------
// Kernel 1: partial column sums over the node dimension.
// grid = B_*S_ blocks, 256 threads (one per feature d). Fully coalesced:
// thread d reads nf[b][n][d] for a contiguous n-chunk.
// ---------------------------------------------------------------------------
__global__ void gcn_colsum_partial(const float* __restrict__ nf,
                                   float* __restrict__ partials) {
    const int chunk = (N_ + S_ - 1) / S_;          // 63
    int bid = blockIdx.x;                          // b * S_ + s
    int b = bid / S_;
    int s = bid % S_;
    int d = threadIdx.x;
    int n0 = s * chunk;
    int n1 = min(N_, n0 + chunk);
    const float* base = nf + (size_t)b * N_ * D_ + d;
    float acc = 0.0f;
    for (int n = n0; n < n1; ++n) acc += base[(size_t)n * D_];
    partials[(size_t)bid * D_ + d] = acc;
}

// ---------------------------------------------------------------------------
// Kernel 2: the whole 3-layer MLP on the 32x256 mean matrix, one workgroup.
// 1024 threads = 32 wave32s; 32 output tiles (2 x 16 of 16x16) per layer,
// one tile per wave. Uses V_WMMA_F32_16X16X4_F32 (exact fp32).
//   A (16x4 f32, 2 VGPRs): lane L -> M = L%16, K = (L/16)*2 + v
//   B (4x16 f32, 2 VGPRs): lane L -> N = L%16, K = (L/16)*2 + v
//   C/D (16x16 f32, 8 VGPRs): VGPR r -> M = r + 8*(L/16), N = L%16
// H lives in LDS ping-pong buffers (padded stride); W/bias read from global
// (tiny, L2-resident).
// ---------------------------------------------------------------------------
__global__ __launch_bounds__(1024) void gcn_mlp_wmma(
        const float* __restrict__ partials,
        const float* __restrict__ Ws,
        const float* __restrict__ bs,
        float* __restrict__ v3) {
    extern __shared__ float H[];                   // 2 * B_ * HSTRIDE floats
    float* Hbuf[2] = { H, H + B_ * HSTRIDE };

    const int tid = threadIdx.x;

    // Reduce S_ partials -> mean matrix H[0] (B_ x D_, padded rows)
    for (int idx = tid; idx < B_ * D_; idx += 1024) {
        int b = idx >> 8;            // / D_
        int d = idx & (D_ - 1);      // % D_
        float acc = 0.0f;
        for (int s = 0; s < S_; ++s)
            acc += partials[((size_t)(b * S_ + s)) * D_ + d];
        Hbuf[0][b * HSTRIDE + d] = acc * (1.0f / (float)N_);
    }
    __syncthreads();

    const int wave = tid >> 5;
    const int lane = tid & 31;
    const int lo = lane & 15;        // M/N index within tile
    const int hi = lane >> 4;        // K sub-group select
    const int m0 = (wave & 1) * 16;  // 2 row tiles (B_ = 32)
    const int n0 = (wave >> 1) * 16; // 16 col tiles (D_ = 256)

    int cur = 0;
    for (int l = 0; l < L_; ++l) {
        const float* W = Ws + (size_t)l * D_ * D_;
        const float* hrow = &Hbuf[cur][(m0 + lo) * HSTRIDE];
        const float* wcol = W + n0 + lo;

        v8f c = {};
        for (int k0 = 0; k0 < D_; k0 += 4) {
            int ka = k0 + hi * 2;
            v2f a, bm;
            a.x  = hrow[ka + 0];
            a.y  = hrow[ka + 1];
            bm.x = wcol[(size_t)(ka + 0) * D_];
            bm.y = wcol[(size_t)(ka + 1) * D_];
            c = __builtin_amdgcn_wmma_f32_16x16x4_f32(
                    /*neg_a=*/false, a, /*neg_b=*/false, bm,
                    /*c_mod=*/(short)0, c, /*reuse_a=*/false, /*reuse_b=*/false);
        }

        const float bias = bs[l * D_ + n0 + lo];
        if (l == L_ - 1) {
            // final layer: no relu, write v3 to global scratch
            #pragma unroll
            for (int r = 0; r < 8; ++r) {
                int m = m0 + r + hi * 8;
                v3[(size_t)m * D_ + n0 + lo] = c[r] + bias;
            }
        } else {
            float* Hn = Hbuf[cur ^ 1];
            #pragma unroll
            for (int r = 0; r < 8; ++r) {
                int m = m0 + r + hi * 8;
                Hn[m * HSTRIDE + n0 + lo] = fmaxf(c[r] + bias, 0.0f);
            }
        }
        cur ^= 1;
        __syncthreads();
    }
}

// ---------------------------------------------------------------------------
// Kernel 3: out1 = nf + broadcast(v3), out2 = nf. float4-vectorized,
// one pass: 16B load nf, 16B (cached) load v3, 32B store.
// ---------------------------------------------------------------------------
__global__ void gcn_broadcast_out(const float* __restrict__ nf,
                                  const float* __restrict__ v3,
                                  float* __restrict__ out) {
    const size_t total4 = (size_t)B_ * N_ * D_ / 4;   // 2,048,000
    size_t i = (size_t)blockIdx.x * blockDim.x + threadIdx.x;
    if (i >= total4) return;

    const float4* nf4 = (const float4*)nf;
    const float4* v34 = (const float4*)v3;
    float4* o1 = (float4*)out;
    float4* o2 = o1 + total4;

    const size_t nd4 = (size_t)N_ * D_ / 4;           // elems per batch (vec4)
    int b  = (int)(i / nd4);
    int d4 = (int)(i % (D_ / 4));

    float4 x = nf4[i];
    float4 v = v34[(size_t)b * (D_ / 4) + d4];
    float4 y;
    y.x = x.x + v.x; y.y = x.y + v.y; y.z = x.z + v.z; y.w = x.w + v.w;
    o1[i] = y;                                        // update_node_feature
    o2[i] = x;                                        // node_feature passthrough
}

// ---------------------------------------------------------------------------
extern "C" void kernel_launch(void* const* d_in, const int* in_sizes, int n_in,
                              void* d_out, int out_size, void* d_ws, size_t ws_size,
                              hipStream_t stream) {
    (void)in_sizes; (void)n_in; (void)out_size; (void)ws_size;

    // setup_inputs order: x (unused by the math), node_feature, Ws, bs
    const float* nf = (const float*)d_in[1];
    const float* Ws = (const float*)d_in[2];
    const float* bs = (const float*)d_in[3];
    float* out = (float*)d_out;

    float* partials = (float*)d_ws;                       // B_*S_*D_ floats (512 KB)
    float* v3       = partials + (size_t)B_ * S_ * D_;    // B_*D_ floats   (32 KB)

    gcn_colsum_partial<<<B_ * S_, D_, 0, stream>>>(nf, partials);

    const size_t shmem = (size_t)2 * B_ * HSTRIDE * sizeof(float);  // ~66.6 KB
    gcn_mlp_wmma<<<1, 1024, shmem, stream>>>(partials, Ws, bs, v3);

    const size_t total4 = (size_t)B_ * N_ * D_ / 4;
    gcn_broadcast_out<<<(unsigned)((total4 + 255) / 256), 256, 0, stream>>>(nf, v3, out);
}